// RecursiveWeight_66305705115675
// MI455X (gfx1250) — compile-verified
//
#include <hip/hip_runtime.h>

// Problem constants (match reference module defaults)
#define DDIM   4096          // dimension_size
#define TDIM   16            // time steps (== WMMA M)
#define RREF   4             // recursive refs
#define KHARM  8             // harmonic components
#define NSTEPS 8             // recursion depth (max; guarded by device scalar)
#define KCH    (DDIM / 4)    // k-chunks of 4 per WMMA  -> 1024
#define NTILES (DDIM / 16)   // 16-wide output column tiles -> 256

#define SPLIT  4             // K-split ways per output tile
#define TPB    2             // N tiles per block
#define KCHS   (KCH / SPLIT) // k-chunks per wave -> 256
#define NBLK   (NTILES / TPB)            // 128 blocks
#define APACK_FLOATS (TDIM * DDIM)       // 65536 floats = 256 KB
#define RED_FLOATS   (TPB * (SPLIT - 1) * 256)  // partial-tile buffer

typedef __attribute__((ext_vector_type(2))) float v2f;
typedef __attribute__((ext_vector_type(8))) float v8f;

// ---------------------------------------------------------------------------
// Kernel 1: Bpack[nt][kc][lane][j] = sum_r w[r] * trans[r][N][K]
//   with N = nt*16 + (lane&15), K = 4*kc + 2*(lane>>4) + j
// Exactly the V_WMMA_F32_16X16X4_F32 B-fragment layout -> the GEMM loads B
// with one coalesced global_load_b64 per lane.
// ---------------------------------------------------------------------------
__global__ __launch_bounds__(256)
void pack_b_kernel(const float* __restrict__ trans,
                   const float* __restrict__ contrib_w,
                   float* __restrict__ bpack) {
    unsigned idx  = blockIdx.x * 256u + threadIdx.x;   // 0 .. D*D-1
    unsigned j    = idx & 1u;
    unsigned lane = (idx >> 1) & 31u;
    unsigned kc   = (idx >> 6) & (KCH - 1u);
    unsigned nt   = idx >> 16;
    unsigned K    = 4u * kc + 2u * (lane >> 4) + j;
    unsigned N    = nt * 16u + (lane & 15u);
    float acc = 0.0f;
#pragma unroll
    for (int r = 0; r < RREF; ++r)
        acc += contrib_w[r] *
               trans[(size_t)r * DDIM * DDIM + (size_t)N * DDIM + K];
    bpack[idx] = acc;
}

// ---------------------------------------------------------------------------
// Kernel 2: affine[t][d] = codebook[bi][d]*SCALE + base_phase[d]
//                        + sum_k amp[k][d]*sin(freq[k]*t + off[k]) + err[d]
// This is also the depth-0 state W0.
// ---------------------------------------------------------------------------
__global__ __launch_bounds__(256)
void affine_kernel(const float* __restrict__ codebook,
                   const int*   __restrict__ base_index,
                   const float* __restrict__ t_arr,
                   const float* __restrict__ base_phase,
                   const float* __restrict__ amps,
                   const float* __restrict__ freqs,
                   const float* __restrict__ offs,
                   const float* __restrict__ err,
                   float* __restrict__ affine) {
    unsigned idx = blockIdx.x * 256u + threadIdx.x;    // 0 .. T*D-1
    unsigned t   = idx >> 12;                          // / DDIM
    unsigned d   = idx & (DDIM - 1u);
    float tv = t_arr[t];
    float ph = base_phase[d];
#pragma unroll
    for (int k = 0; k < KHARM; ++k)
        ph += amps[(size_t)k * DDIM + d] * sinf(tv * freqs[k] + offs[k]);
    float base = codebook[(size_t)(*base_index) * DDIM + d] * 1.0f; // SCALE=1
    affine[idx] = base + ph + err[d];
}

// ---------------------------------------------------------------------------
// Kernel 3: one recursion step.
//   Wout[t][d] = affine[t][d] + clamp(base_delta[d]*dfac*adapt[d], +-1000)
//              + sum_e Bcomb[d][e] * Win[t][e]
// Block = 8 waves = TPB(2) N-tiles x SPLIT(4) K-splits; 128 blocks cover all
// 256 tiles. W is staged into LDS in A-fragment layout with CDNA5 async
// global->LDS b64 copies (ASYNCcnt), then each wave runs 256 x
// v_wmma_f32_16x16x4_f32 over its K quarter; partials reduced via LDS.
// ---------------------------------------------------------------------------
__global__ __launch_bounds__(256)
void step_kernel(const float* __restrict__ Win,
                 const float* __restrict__ Bpack,
                 const float* __restrict__ affine,
                 const float* __restrict__ base_delta,
                 const float* __restrict__ adapt,
                 const int*   __restrict__ rec_depth,
                 float* __restrict__ Wout,
                 int depth, float dfactor) {
    extern __shared__ float Apack[];   // [APACK_FLOATS] + [RED_FLOATS]
    float* red = Apack + APACK_FLOATS;
    const int tid  = threadIdx.x;
    const int lane = tid & 31;
    const int wave = tid >> 5;

    // Pass-through when this step exceeds requested recursion depth
    // (uniform branch across the whole grid: no barrier divergence,
    //  EXEC stays all-ones on the WMMA path).
    if (depth > *rec_depth) {
        for (int i = blockIdx.x * 256 + tid; i < TDIM * DDIM; i += NBLK * 256)
            Wout[i] = Win[i];
        return;
    }

    // Async pack: Win (row-major [t][e]) -> LDS A fragments, b64 per pair.
    // Pair p = kc*32 + l holds W[t = l&15, e = 4kc + 2*(l>>4) + {0,1}]
    // at LDS float offset 2p. global_load_async_to_lds_b64 takes the per-lane
    // LDS byte address in VDST and the 64-bit global address in VADDR.
    for (int p = tid; p < (TDIM * DDIM) / 2; p += 256) {
        int l  = p & 31;
        int kc = p >> 5;
        int t  = l & 15;
        int e  = 4 * kc + 2 * (l >> 4);
        const float* g = Win + (size_t)t * DDIM + e;
        unsigned lds_addr = (unsigned)(uintptr_t)(const void*)(Apack + 2 * p);
        asm volatile("global_load_async_to_lds_b64 %0, %1, off"
                     :: "v"(lds_addr), "v"(g) : "memory");
    }
    asm volatile("s_wait_asynccnt 0x0" ::: "memory");
    __syncthreads();

    const int tile  = wave >> 2;                 // 0..TPB-1
    const int split = wave & (SPLIT - 1);        // 0..SPLIT-1
    const int nt    = blockIdx.x * TPB + tile;   // this tile's N index
    const int kc0   = split * KCHS;
    const float* __restrict__ bp = Bpack + (size_t)nt * (KCH * 64) + lane * 2;
    const float* __restrict__ ap = Apack + lane * 2;

    v8f acc = {};
#pragma unroll 4
    for (int kc = kc0; kc < kc0 + KCHS; ++kc) {
        v2f a = *(const v2f*)(ap + kc * 64);           // ds_load_b64, conflict-free
        v2f b = *(const v2f*)(bp + (size_t)kc * 64);   // global_load_b64, coalesced
        __builtin_prefetch(bp + (size_t)(kc + 16) * 64, 0, 1);
        acc = __builtin_amdgcn_wmma_f32_16x16x4_f32(
            /*neg_a=*/false, a, /*neg_b=*/false, b,
            /*c_mod=*/(short)0, acc, /*reuse_a=*/false, /*reuse_b=*/false);
    }

    // K-split reduction through LDS: splits 1..3 park partial tiles,
    // split 0 accumulates them and runs the epilogue.
    if (split != 0)
        *(v8f*)(red + ((tile * (SPLIT - 1) + (split - 1)) * 256) + lane * 8) = acc;
    __syncthreads();
    if (split != 0)
        return;
#pragma unroll
    for (int s = 0; s < SPLIT - 1; ++s)
        acc += *(const v8f*)(red + ((tile * (SPLIT - 1) + s) * 256) + lane * 8);

    // Epilogue: C layout -> VGPR i, lane L<16: (M=i, N=L); lane L+16: (M=i+8).
    const int half = lane >> 4;
    const int dcol = nt * 16 + (lane & 15);
    float dl = base_delta[dcol] * dfactor * adapt[dcol];
    dl = fminf(fmaxf(dl, -1000.0f), 1000.0f);
#pragma unroll
    for (int i = 0; i < 8; ++i) {
        int m = i + half * 8;
        size_t o = (size_t)m * DDIM + dcol;
        Wout[o] = affine[o] + dl + acc[i];
    }
}

// ---------------------------------------------------------------------------
extern "C" void kernel_launch(void* const* d_in, const int* in_sizes, int n_in,
                              void* d_out, int out_size, void* d_ws, size_t ws_size,
                              hipStream_t stream) {
    const float* codebook = (const float*)d_in[0];
    const float* t_arr    = (const float*)d_in[1];
    const float* trans    = (const float*)d_in[2];
    const float* contrib  = (const float*)d_in[3];
    const float* basep    = (const float*)d_in[4];
    const float* amps     = (const float*)d_in[5];
    const float* freqs    = (const float*)d_in[6];
    const float* offs     = (const float*)d_in[7];
    const float* err      = (const float*)d_in[8];
    const float* bdelta   = (const float*)d_in[9];
    const float* adapt    = (const float*)d_in[10];
    const int*   bindex   = (const int*)d_in[11];
    const int*   rdepth   = (const int*)d_in[12];

    float* bpack  = (float*)d_ws;                          // 64 MB (L2-resident)
    float* affine = bpack  + (size_t)DDIM * DDIM;          // 256 KB
    float* ping   = affine + (size_t)TDIM * DDIM;          // 256 KB
    float* pong   = ping   + (size_t)TDIM * DDIM;          // 256 KB

    pack_b_kernel<<<(DDIM * DDIM) / 256, 256, 0, stream>>>(trans, contrib, bpack);
    affine_kernel<<<(TDIM * DDIM) / 256, 256, 0, stream>>>(
        codebook, bindex, t_arr, basep, amps, freqs, offs, err, affine);

    const size_t lds_bytes = (size_t)(APACK_FLOATS + RED_FLOATS) * sizeof(float);
    const float* cur = affine;                             // depth-0 state == affine
    float df = 1.0f;
    for (int step = 1; step <= NSTEPS; ++step) {
        df *= 0.95f;                                       // 0.95^step
        float* out = (step == NSTEPS) ? (float*)d_out
                                      : ((step & 1) ? ping : pong);
        step_kernel<<<NBLK, 256, lds_bytes, stream>>>(
            cur, bpack, affine, bdelta, adapt, rdepth, out, step, df);
        cur = out;
    }
}